// DATKLoss_43739946942559
// MI455X (gfx1250) — compile-verified
//
#include <hip/hip_runtime.h>
#include <hip/hip_bf16.h>
#include <math.h>

#define BATCH   64
#define NANCH   8400
#define NCLS    80
#define STRIDE  84          // floats per anchor
#define TILE    128         // anchors per block in kernel A
#define GB      66          // ceil(8400/128)
#define GB2     8           // chunks per batch in kernel B
#define EPSV    1e-9f

// -log(max(x, EPS)) * x
__device__ __forceinline__ float plog(float x) {
    return -__logf(fmaxf(x, EPSV)) * x;
}

__device__ __forceinline__ void async_ld16(unsigned laddr, const void* gptr) {
    unsigned long gaddr = (unsigned long)(uintptr_t)gptr;
    asm volatile("global_load_async_to_lds_b128 %0, %1, off"
                 :: "v"(laddr), "v"(gaddr) : "memory");
}

// ---------------------------------------------------------------------------
// Kernel A: stage tile to LDS via CDNA5 async copies, per-anchor math,
// pre-bishop mask, per-batch maxima via wave-reduce + uint atomicMax.
// ---------------------------------------------------------------------------
__global__ __launch_bounds__(TILE) void kA(const float* __restrict__ pred,
                                           const float* __restrict__ label,
                                           float* __restrict__ iouA,
                                           float* __restrict__ confA,
                                           float* __restrict__ logitA,
                                           unsigned* __restrict__ maxIU,
                                           unsigned* __restrict__ maxCU) {
    __shared__ float buf[TILE * STRIDE];          // 43,008 bytes

    const int tid  = threadIdx.x;
    const int bb   = blockIdx.x / GB;
    const int tile = blockIdx.x % GB;
    const int t0   = tile * TILE;
    const int nA   = min(TILE, NANCH - t0);

    // ---- async global -> LDS staging (coalesced b128 per lane) ----
    const float* gbase = pred + ((size_t)bb * NANCH + t0) * STRIDE; // 16B aligned
    const unsigned lds_base = (unsigned)(uintptr_t)(&buf[0]);       // low 32b = LDS offset
    if (nA == TILE) {
        // fast path: full tile, exactly 21 chunks per thread, branch-free
#pragma unroll
        for (int j = 0; j < STRIDE / 4; ++j) {
            const int c = j * TILE + tid;
            async_ld16(lds_base + (unsigned)(c * 16), gbase + (size_t)c * 4);
        }
    } else {
        const int nChunks = nA * (STRIDE / 4);
        for (int c = tid; c < nChunks; c += TILE)
            async_ld16(lds_base + (unsigned)(c * 16), gbase + (size_t)c * 4);
    }
    asm volatile("s_wait_asynccnt 0x0" ::: "memory");
    __syncthreads();

    const float4 lb = ((const float4*)label)[bb];   // lx1, ly1, lx2, ly2

    float mi = 0.0f, mc = 0.0f;                     // masked maxima candidates
    if (tid < nA) {
        const float*  ap = buf + tid * STRIDE;
        const float4  bx = *(const float4*)ap;      // cx, cy, w, h
        const float4* c4 = (const float4*)(ap + 4); // 20 x float4 of class confs

        const float px1 = bx.x - bx.z * 0.5f, px2 = bx.x + bx.z * 0.5f;
        const float py1 = bx.y - bx.w * 0.5f, py2 = bx.y + bx.w * 0.5f;
        const float xk1 = fmaxf(lb.x, px1), yk1 = fmaxf(lb.y, py1);
        const float xk2 = fminf(lb.z, px2), yk2 = fminf(lb.w, py2);
        const float inter = ((yk2 > yk1) && (xk2 > xk1))
                          ? (xk2 - xk1) * (yk2 - yk1) : 0.0f;
        const float uni = (lb.z - lb.x) * (lb.w - lb.y)
                        + (px2 - px1) * (py2 - py1) - inter;
        const float iou = inter / uni;

        // single pass: class sum, keep-masked max (class 0 excluded), conf
        float conf, sa, mxk;
        {
            float4 v = c4[0];
            conf = v.x;
            sa   = v.x + v.y + v.z + v.w;
            mxk  = fmaxf(fmaxf(v.y, v.z), v.w);
        }
#pragma unroll
        for (int j = 1; j < NCLS / 4; ++j) {
            float4 v = c4[j];
            sa  += (v.x + v.y) + (v.z + v.w);
            mxk  = fmaxf(mxk, fmaxf(fmaxf(v.x, v.y), fmaxf(v.z, v.w)));
        }

        const bool m0 = (iou > 0.45f) && (conf > 0.25f) && (conf > mxk * 0.9f);

        // logits only matter where the final mask can be true (final ⊆ m0)
        float logits = 0.0f;
        if (m0) {
            float p2n;
            {
                float4 v = c4[0];                    // class 0 term is zeroed by keep
                p2n = plog(v.y) + plog(v.z) + plog(v.w);
            }
#pragma unroll
            for (int j = 1; j < NCLS / 4; ++j) {
                float4 v = c4[j];
                p2n += (plog(v.x) + plog(v.y)) + (plog(v.z) + plog(v.w));
            }
            const float sadj = sa - fmaxf(sa + 1e-6f - 1.0f, 0.0f);
            const float last = 1.0f - sadj;
            const float csum = (sa - conf) + last;   // conf_others.sum()
            p2n += plog(last);
            logits = -__logf(fmaxf(csum, EPSV)) + p2n / (csum + EPSV);
        }

        const size_t gi = (size_t)bb * NANCH + t0 + tid;
        iouA[gi]   = iou;
        confA[gi]  = m0 ? conf : -conf;              // pack mask0 into sign
        logitA[gi] = logits;
        if (m0) { mi = iou; mc = conf; }
    }

    // wave32 max-reduce, one atomic per wave (uint compare == float compare, x>=0)
#pragma unroll
    for (int off = 16; off; off >>= 1) {
        mi = fmaxf(mi, __shfl_xor(mi, off, 32));
        mc = fmaxf(mc, __shfl_xor(mc, off, 32));
    }
    if ((tid & 31) == 0) {
        atomicMax(&maxIU[bb], __float_as_uint(mi));
        atomicMax(&maxCU[bb], __float_as_uint(mc));
    }
}

// ---------------------------------------------------------------------------
// Kernel B: bishop refinement + deterministic per-chunk partial sums.
// ---------------------------------------------------------------------------
__global__ __launch_bounds__(256) void kB(const float* __restrict__ iouA,
                                          const float* __restrict__ confA,
                                          const float* __restrict__ logitA,
                                          const unsigned* __restrict__ maxIU,
                                          const unsigned* __restrict__ maxCU,
                                          float* __restrict__ partials) {
    __shared__ float red[24];

    const int tid = threadIdx.x;
    const int bb  = blockIdx.x / GB2;
    const int g   = blockIdx.x % GB2;

    const float mxI = __uint_as_float(maxIU[bb]) * 0.5f;
    const float mxC = __uint_as_float(maxCU[bb]) * 0.5f;

    float cnt = 0.0f, sA = 0.0f, sB = 0.0f;
    for (int i = g * 256 + tid; i < NANCH; i += GB2 * 256) {
        const size_t gi  = (size_t)bb * NANCH + i;
        const float  cm  = confA[gi];
        const float  iou = iouA[gi];
        const bool   m   = (cm > 0.0f) && (cm > mxC) && (iou > mxI);
        if (m) {
            cnt += 1.0f;
            sA  += logitA[gi] * iou;
            sB  += cm * iou;
        }
    }
#pragma unroll
    for (int off = 16; off; off >>= 1) {
        cnt += __shfl_xor(cnt, off, 32);
        sA  += __shfl_xor(sA,  off, 32);
        sB  += __shfl_xor(sB,  off, 32);
    }
    const int wid = tid >> 5;
    if ((tid & 31) == 0) { red[wid] = cnt; red[8 + wid] = sA; red[16 + wid] = sB; }
    __syncthreads();
    if (tid == 0) {
        float c = 0.0f, a = 0.0f, s = 0.0f;
#pragma unroll
        for (int w = 0; w < 8; ++w) { c += red[w]; a += red[8 + w]; s += red[16 + w]; }
        float* p = partials + ((size_t)bb * GB2 + g) * 3;
        p[0] = c; p[1] = a; p[2] = s;
    }
}

// ---------------------------------------------------------------------------
// Kernel C: final per-batch divides.  d_out = [loss(64) | loss_value(64)].
// ---------------------------------------------------------------------------
__global__ void kC(const float* __restrict__ partials, float* __restrict__ out) {
    const int b = threadIdx.x;
    if (b < BATCH) {
        float c = EPSV, a = 0.0f, s = 0.0f;   // mask_cnt = sum(mf) + EPS
#pragma unroll
        for (int g = 0; g < GB2; ++g) {
            const float* p = partials + ((size_t)b * GB2 + g) * 3;
            c += p[0]; a += p[1]; s += p[2];
        }
        out[b]         = a / c;               // loss
        out[BATCH + b] = s / c + EPSV;        // loss_value
    }
}

// ---------------------------------------------------------------------------
extern "C" void kernel_launch(void* const* d_in, const int* in_sizes, int n_in,
                              void* d_out, int out_size, void* d_ws, size_t ws_size,
                              hipStream_t stream) {
    (void)in_sizes; (void)n_in; (void)out_size; (void)ws_size;
    const float* pred  = (const float*)d_in[0];   // (64, 8400, 84) f32
    const float* label = (const float*)d_in[1];   // (64, 4) f32
    float* out = (float*)d_out;                   // 128 f32
    float* ws  = (float*)d_ws;

    unsigned* maxIU    = (unsigned*)ws;               // 64
    unsigned* maxCU    = (unsigned*)(ws + 64);        // 64
    float*    partials = ws + 128;                    // 64*8*3
    float*    iouA     = ws + 2048;                   // B*N
    float*    confA    = iouA  + (size_t)BATCH * NANCH;
    float*    logitA   = confA + (size_t)BATCH * NANCH;

    hipMemsetAsync(ws, 0, 128 * sizeof(float), stream);   // zero maxI/maxC
    kA<<<dim3(BATCH * GB),  dim3(TILE), 0, stream>>>(pred, label, iouA, confA,
                                                     logitA, maxIU, maxCU);
    kB<<<dim3(BATCH * GB2), dim3(256),  0, stream>>>(iouA, confA, logitA,
                                                     maxIU, maxCU, partials);
    kC<<<dim3(1), dim3(BATCH), 0, stream>>>(partials, out);
}